// GCNN_desc_pool_26688926777485
// MI455X (gfx1250) — compile-verified
//
#include <hip/hip_runtime.h>
#include <hip/hip_bf16.h>
#include <math.h>

typedef float v8f __attribute__((ext_vector_type(8)));
typedef float v2f __attribute__((ext_vector_type(2)));
typedef int   v4i_gcc __attribute__((vector_size(16)));   // matches builtin param pointee

#define F_DIM    1024
#define D_DIM    128
#define DESC_DIM 80
#define L_DIM    2048
#define CONV_CHUNK 128   // L rows per conv block -> gridDim.x = 16

#define BM 128
#define BN 64
#define BK 32
#define LDA_P  36   // A tile LDS pitch: 16B-aligned rows, conflict-free frag reads
#define LDB_PT 36   // transposed-B tile LDS pitch

#define GLOBAL_AS __attribute__((address_space(1)))
#define LDS_AS    __attribute__((address_space(3)))

#if defined(__has_builtin)
#  if __has_builtin(__builtin_amdgcn_global_load_async_to_lds_b128)
#    define HAVE_ASYNC_LDS 1
#  endif
#  if __has_builtin(__builtin_amdgcn_s_wait_asynccnt)
#    define ASYNC_WAIT() __builtin_amdgcn_s_wait_asynccnt(0)
#  elif defined(HAVE_ASYNC_LDS)
#    define ASYNC_WAIT() asm volatile("s_wait_asynccnt 0" ::: "memory")
#  endif
#endif

__device__ __forceinline__ float leaky(float v) { return v > 0.0f ? v : 0.01f * v; }

__device__ __forceinline__ void fadd_atomic(float* p, float v) {
    (void)__hip_atomic_fetch_add(p, v, __ATOMIC_RELAXED, __HIP_MEMORY_SCOPE_AGENT);
}

// monotone float <-> uint encoding for atomic max on possibly-negative floats
__device__ __forceinline__ unsigned enc_f(float f) {
    unsigned u = __float_as_uint(f);
    return (u & 0x80000000u) ? ~u : (u | 0x80000000u);
}
__device__ __forceinline__ float dec_f(unsigned k) {
    unsigned u = (k & 0x80000000u) ? (k & 0x7FFFFFFFu) : ~k;
    return __uint_as_float(u);
}

// ---------------------------------------------------------------------------
// Fill
// ---------------------------------------------------------------------------
__global__ void k_fill(float* __restrict__ p, long n, float v) {
    long i = (long)blockIdx.x * blockDim.x + threadIdx.x;
    if (i < n) p[i] = v;
}

// ---------------------------------------------------------------------------
// fp32 WMMA GEMM: C[M,N] = A[M,K] * W[K,N]  (row-major everywhere)
// block: 256 thr = 8 waves; block tile 128x64; wave tile 32x32 (2x2 of 16x16)
// A tile staged via async global->LDS (when available); B staged transposed.
// ---------------------------------------------------------------------------
__global__ __launch_bounds__(256) void k_gemm_wmma(
    const float* __restrict__ A, const float* __restrict__ W,
    float* __restrict__ C, int M, int K, int Nn)
{
    __shared__ float sA[BM * LDA_P];     // [128][36] row-major (m, k)
    __shared__ float sBt[BN * LDB_PT];   // [64][36]  transposed (n, k)

    const int tid  = threadIdx.x;
    const int lane = tid & 31;
    const int wave = tid >> 5;
    const int half = lane >> 4;   // 0: K-pair {0,1}, 1: K-pair {2,3}
    const int l16  = lane & 15;
    const int wm   = (wave >> 1) * 32;
    const int wn   = (wave & 1) * 32;
    const int bm   = blockIdx.y * BM;
    const int bn   = blockIdx.x * BN;

    v8f acc[2][2];
    #pragma unroll
    for (int i = 0; i < 2; ++i)
        #pragma unroll
        for (int j = 0; j < 2; ++j)
            acc[i][j] = (v8f)0.0f;

    for (int k0 = 0; k0 < K; k0 += BK) {
        // ---- stage A tile: 128x32 floats (1024 xB128, 4 per thread) ----
        #pragma unroll
        for (int i = 0; i < 4; ++i) {
            int idx = tid + i * 256;         // float4 index 0..1023
            int row = idx >> 3;              // 8 float4 per row
            int c4  = (idx & 7) << 2;
            int gm  = bm + row;
            float* ld = &sA[row * LDA_P + c4];
            if (gm < M) {
                const float* gp = A + (size_t)gm * K + k0 + c4;
#ifdef HAVE_ASYNC_LDS
                __builtin_amdgcn_global_load_async_to_lds_b128(
                    (GLOBAL_AS v4i_gcc*)gp, (LDS_AS v4i_gcc*)ld, 0, 0);
#else
                float4 v = *(const float4*)gp;
                ld[0] = v.x; ld[1] = v.y; ld[2] = v.z; ld[3] = v.w;
#endif
            } else {
                ld[0] = 0.f; ld[1] = 0.f; ld[2] = 0.f; ld[3] = 0.f;
            }
        }
        // ---- stage B tile transposed: sBt[n][k] = W[k0+k][bn+n] ----
        #pragma unroll
        for (int i = 0; i < 2; ++i) {
            int idx  = tid + i * 256;        // 0..511
            int krow = idx >> 4;             // 16 float4 per row
            int n0   = (idx & 15) << 2;
            float4 v = *(const float4*)(W + (size_t)(k0 + krow) * Nn + bn + n0);
            sBt[(n0 + 0) * LDB_PT + krow] = v.x;
            sBt[(n0 + 1) * LDB_PT + krow] = v.y;
            sBt[(n0 + 2) * LDB_PT + krow] = v.z;
            sBt[(n0 + 3) * LDB_PT + krow] = v.w;
        }
        // prefetch next W tile into cache (global_prefetch_b8)
        if (k0 + BK < K)
            __builtin_prefetch(W + (size_t)(k0 + BK + (tid >> 4)) * Nn + bn + ((tid & 15) << 2), 0, 0);
#ifdef HAVE_ASYNC_LDS
        ASYNC_WAIT();
#endif
        __syncthreads();

        #pragma unroll
        for (int kk = 0; kk < BK; kk += 4) {
            const int kb = kk + 2 * half;
            v2f afrag[2], bfrag[2];
            #pragma unroll
            for (int i = 0; i < 2; ++i) {
                const float* pa = &sA[(wm + i * 16 + l16) * LDA_P + kb];
                afrag[i][0] = pa[0];
                afrag[i][1] = pa[1];
            }
            #pragma unroll
            for (int j = 0; j < 2; ++j) {
                const float* pb = &sBt[(wn + j * 16 + l16) * LDB_PT + kb];
                bfrag[j][0] = pb[0];
                bfrag[j][1] = pb[1];
            }
            #pragma unroll
            for (int i = 0; i < 2; ++i)
                #pragma unroll
                for (int j = 0; j < 2; ++j)
                    acc[i][j] = __builtin_amdgcn_wmma_f32_16x16x4_f32(
                        false, afrag[i], false, bfrag[j],
                        (short)0, acc[i][j], false, false);
        }
        __syncthreads();
    }

    // store: VGPR v -> rows {v, v+8} by lane half; cols = l16
    #pragma unroll
    for (int i = 0; i < 2; ++i)
        #pragma unroll
        for (int j = 0; j < 2; ++j)
            #pragma unroll
            for (int v = 0; v < 8; ++v) {
                int row = bm + wm + i * 16 + v + half * 8;
                int col = bn + wn + j * 16 + l16;
                if (row < M) C[(size_t)row * Nn + col] = acc[i][j][v];
            }
}

// ---------------------------------------------------------------------------
// Graph pieces
// ---------------------------------------------------------------------------
__global__ void k_deg(const int* __restrict__ ei, float* __restrict__ deg, int E) {
    int e = blockIdx.x * blockDim.x + threadIdx.x;
    if (e < E) fadd_atomic(&deg[ei[E + e]], 1.0f);
}

__global__ void k_count(const int* __restrict__ batch, float* __restrict__ counts, int N) {
    int i = blockIdx.x * blockDim.x + threadIdx.x;
    if (i < N) fadd_atomic(&counts[batch[i]], 1.0f);
}

// dinv = rsqrt(deg); agg init = self-loop term h[i]*dinv^2
__global__ void k_selfagg(const float* __restrict__ h, const float* __restrict__ deg,
                          float* __restrict__ dinv, float* __restrict__ agg, long NF) {
    long idx = (long)blockIdx.x * blockDim.x + threadIdx.x;
    if (idx >= NF) return;
    long i = idx >> 10;
    float d = rsqrtf(deg[i]);
    agg[idx] = h[idx] * d * d;
    if ((idx & 1023) == 0) dinv[i] = d;
}

// one block (256 thr) per edge: agg[dst] += h[src] * dinv[src]*dinv[dst]
__global__ __launch_bounds__(256) void k_scatter(
    const float* __restrict__ h, const float* __restrict__ dinv,
    const int* __restrict__ ei, float* __restrict__ agg, int E)
{
    int e = blockIdx.x;
    int src = ei[e], dst = ei[E + e];
    float norm = dinv[src] * dinv[dst];
    const float* hs = h + (size_t)src * F_DIM;
    float* ad = agg + (size_t)dst * F_DIM;
    for (int f = threadIdx.x; f < F_DIM; f += 256)
        fadd_atomic(&ad[f], hs[f] * norm);
}

// v = leaky(agg + bias); pooled[batch[i]] += v
__global__ void k_finalize_pool(const float* __restrict__ agg, const float* __restrict__ bg,
                                const int* __restrict__ batch, float* __restrict__ pooled, long NF)
{
    long idx = (long)blockIdx.x * blockDim.x + threadIdx.x;
    if (idx >= NF) return;
    long i = idx >> 10;
    int  f = (int)(idx & 1023);
    float v = leaky(agg[idx] + bg[f]);
    fadd_atomic(&pooled[(size_t)batch[i] * F_DIM + f], v);
}

// feat = leaky((pooled/count) @ Wfc + b); one block per graph, thread = out col
__global__ __launch_bounds__(D_DIM) void k_fc(
    const float* __restrict__ pooled, const float* __restrict__ counts,
    const float* __restrict__ Wfc, const float* __restrict__ bfc,
    float* __restrict__ combXT, int segoff)
{
    int b = blockIdx.x, d = threadIdx.x;
    float cinv = 1.0f / fmaxf(counts[b], 1.0f);
    const float* pb = pooled + (size_t)b * F_DIM;
    float acc = 0.0f;
    for (int f = 0; f < F_DIM; ++f)
        acc += pb[f] * Wfc[(size_t)f * D_DIM + d];
    combXT[(size_t)b * (2 * D_DIM) + segoff + d] = leaky(acc * cinv + bfc[d]);
}

// ---------------------------------------------------------------------------
// Conv1d(k=1) + LeakyReLU + max over L. grid = (L/CONV_CHUNK, B), block = 128
// ---------------------------------------------------------------------------
__global__ __launch_bounds__(D_DIM) void k_conv_max(
    const float* __restrict__ m, const float* __restrict__ Wc,
    const float* __restrict__ bc, unsigned* __restrict__ keys)
{
    __shared__ float sW[DESC_DIM * D_DIM];   // 40 KB
    __shared__ float sRows[32 * DESC_DIM];   // 10 KB
    const int d = threadIdx.x;
    const int b = blockIdx.y;
    const int lbase = blockIdx.x * CONV_CHUNK;

    for (int i = d; i < DESC_DIM * D_DIM; i += D_DIM) sW[i] = Wc[i];
    const float bias = bc[d];
    __syncthreads();

    float best = -INFINITY;
    for (int l0 = lbase; l0 < lbase + CONV_CHUNK; l0 += 32) {
        __syncthreads();
        for (int i = d; i < 32 * DESC_DIM; i += D_DIM) {
            int r = i / DESC_DIM, c = i - r * DESC_DIM;
            sRows[i] = m[((size_t)b * L_DIM + l0 + r) * DESC_DIM + c];
        }
        __syncthreads();
        for (int r = 0; r < 32; ++r) {
            const float* row = &sRows[r * DESC_DIM];
            float acc = bias;
            #pragma unroll 8
            for (int k = 0; k < DESC_DIM; ++k)
                acc += row[k] * sW[k * D_DIM + d];
            best = fmaxf(best, leaky(acc));
        }
    }
    atomicMax(&keys[(size_t)b * D_DIM + d], enc_f(best));
}

// ---------------------------------------------------------------------------
// Final: out[b] = concat(x, xt, m1s, m1f, m2s, m2f) @ W_final + b_final
// ---------------------------------------------------------------------------
__global__ __launch_bounds__(256) void k_final(
    const float* __restrict__ combXT, const unsigned* __restrict__ keys,
    const float* __restrict__ Wf, const float* __restrict__ bf,
    float* __restrict__ out, int B)
{
    __shared__ float red[256];
    int b = blockIdx.x, tid = threadIdx.x;
    float acc = 0.0f;
    for (int j = tid; j < 6 * D_DIM; j += 256) {
        float v;
        if (j < 2 * D_DIM) {
            v = combXT[(size_t)b * (2 * D_DIM) + j];
        } else {
            int jj = j - 2 * D_DIM;
            int seg = jj >> 7, d = jj & (D_DIM - 1);
            v = dec_f(keys[((size_t)seg * B + b) * D_DIM + d]);
        }
        acc += v * Wf[j];
    }
    red[tid] = acc;
    __syncthreads();
    for (int s = 128; s > 0; s >>= 1) {
        if (tid < s) red[tid] += red[tid + s];
        __syncthreads();
    }
    if (tid == 0) out[b] = red[0] + bf[0];
}

// ---------------------------------------------------------------------------
extern "C" void kernel_launch(void* const* d_in, const int* in_sizes, int n_in,
                              void* d_out, int out_size, void* d_ws, size_t ws_size,
                              hipStream_t stream)
{
    const float* pro_x[2]  = {(const float*)d_in[0], (const float*)d_in[3]};
    const int*   pro_ei[2] = {(const int*)d_in[1],   (const int*)d_in[4]};
    const int*   pro_b[2]  = {(const int*)d_in[2],   (const int*)d_in[5]};
    const float* mas[4]    = {(const float*)d_in[6], (const float*)d_in[7],
                              (const float*)d_in[8], (const float*)d_in[9]};
    const float* Wg[2]  = {(const float*)d_in[10], (const float*)d_in[14]};
    const float* bg[2]  = {(const float*)d_in[11], (const float*)d_in[15]};
    const float* Wfc[2] = {(const float*)d_in[12], (const float*)d_in[16]};
    const float* bfc[2] = {(const float*)d_in[13], (const float*)d_in[17]};
    const float* Wm[4]  = {(const float*)d_in[18], (const float*)d_in[20],
                           (const float*)d_in[22], (const float*)d_in[24]};
    const float* bm[4]  = {(const float*)d_in[19], (const float*)d_in[21],
                           (const float*)d_in[23], (const float*)d_in[25]};
    const float* Wfinal = (const float*)d_in[26];
    const float* bfinal = (const float*)d_in[27];
    float* out = (float*)d_out;

    const int N = in_sizes[0] / F_DIM;
    const int E = in_sizes[1] / 2;
    const int B = in_sizes[6] / (L_DIM * DESC_DIM);
    const long NF = (long)N * F_DIM;

    float* ws      = (float*)d_ws;
    float* h       = ws;                         // NF
    float* agg     = h + NF;                     // NF
    float* deg     = agg + NF;                   // N
    float* dinv    = deg + N;                    // N
    float* pooled  = dinv + N;                   // B*F
    float* counts  = pooled + (size_t)B * F_DIM; // B
    float* combXT  = counts + B;                 // B*2D
    unsigned* keys = (unsigned*)(combXT + (size_t)B * 2 * D_DIM); // 4*B*D

    const long nfBlocks = (NF + 255) / 256;

    for (int br = 0; br < 2; ++br) {
        k_fill<<<(N + 255) / 256, 256, 0, stream>>>(deg, N, 1.0f);
        k_fill<<<((long)B * F_DIM + 255) / 256, 256, 0, stream>>>(pooled, (long)B * F_DIM, 0.0f);
        k_fill<<<1, 256, 0, stream>>>(counts, B, 0.0f);

        dim3 ggrid(F_DIM / BN, (N + BM - 1) / BM);
        k_gemm_wmma<<<ggrid, 256, 0, stream>>>(pro_x[br], Wg[br], h, N, F_DIM, F_DIM);

        k_deg<<<(E + 255) / 256, 256, 0, stream>>>(pro_ei[br], deg, E);
        k_count<<<(N + 255) / 256, 256, 0, stream>>>(pro_b[br], counts, N);
        k_selfagg<<<nfBlocks, 256, 0, stream>>>(h, deg, dinv, agg, NF);
        k_scatter<<<E, 256, 0, stream>>>(h, dinv, pro_ei[br], agg, E);
        k_finalize_pool<<<nfBlocks, 256, 0, stream>>>(agg, bg[br], pro_b[br], pooled, NF);
        k_fc<<<B, D_DIM, 0, stream>>>(pooled, counts, Wfc[br], bfc[br], combXT, br * D_DIM);
    }

    k_fill<<<((long)4 * B * D_DIM + 255) / 256, 256, 0, stream>>>((float*)keys, (long)4 * B * D_DIM, 0.0f);
    for (int c = 0; c < 4; ++c)
        k_conv_max<<<dim3(L_DIM / CONV_CHUNK, B), D_DIM, 0, stream>>>(
            mas[c], Wm[c], bm[c], keys + (size_t)c * B * D_DIM);

    k_final<<<B, 256, 0, stream>>>(combXT, keys, Wfinal, bfinal, out, B);
}